// SNN_84791244358143
// MI455X (gfx1250) — compile-verified
//
#include <hip/hip_runtime.h>

typedef __attribute__((ext_vector_type(2))) float v2f;
typedef __attribute__((ext_vector_type(4))) float v4f;
typedef __attribute__((ext_vector_type(8))) float v8f;

#define N_IN    2
#define N_HID   10
#define N_OUT   4
#define N_STEPS 100
#define BATCHSZ 65536
#define BETA    0.5f
#define THRESH  1.0f

__global__ __launch_bounds__(128) void snn_fwd_wmma(
    const float* __restrict__ x,   // [T, B, 2]
    const float* __restrict__ W1,  // [2, 10]
    const float* __restrict__ b1,  // [10]
    const float* __restrict__ W2,  // [10, 4]
    const float* __restrict__ b2,  // [4]
    float* __restrict__ out)       // spk1[T,B,10] | mem1[T,B,10] | spk2[T,B,4] | mem2[T,B,4]
{
    const int lane = threadIdx.x & 31;
    const int wv   = threadIdx.x >> 5;                 // wave in block (0..3)
    const int tile = blockIdx.x * 4 + wv;              // 16 batch rows per wave
    const int b0   = tile * 16;
    const int n    = lane & 15;                        // column (N) index in WMMA B/C/D layout
    const int h    = lane >> 4;                        // lane-half

    // Per-wave LDS slice: memory images of the tile. 448 floats = 1792 B (16B aligned).
    __shared__ float lds[4][448];
    float* s_spk1 = &lds[wv][0];        // 160 floats, image m*10+n
    float* s_mem1 = s_spk1 + 160;       // 160
    float* s_spk2 = s_mem1 + 160;       // 64, image m*4+o
    float* s_mem2 = s_spk2 + 64;        // 64

    // ---- Wave-constant operands -------------------------------------------
    // B1 = W1 padded to 4x16 (K x N). Lane layout: VGPR v + half h -> K = v + 2h.
    v2f B1 = {0.f, 0.f};
    if (h == 0 && n < N_HID) { B1.x = W1[n]; B1.y = W1[N_HID + n]; }
    const float bias1 = (n < N_HID) ? b1[n] : 0.f;
    const float bias2 = (n < N_OUT) ? b2[n] : 0.f;

    // Bias folded into the WMMA C operand: column-constant tile (all 8 rows = bias[n]).
    v8f cbias1, cbias2;
    #pragma unroll
    for (int r = 0; r < 8; ++r) { cbias1[r] = bias1; cbias2[r] = bias2; }

    // B2 chunks: W2 padded to 12x16, split into 3 K=4 slices.
    v2f B2c[3];
    #pragma unroll
    for (int c = 0; c < 3; ++c) {
        const int k0 = 4 * c + 2 * h;
        v2f v = {0.f, 0.f};
        if (n < N_OUT) {
            if (k0     < N_HID) v.x = W2[k0 * N_OUT + n];
            if (k0 + 1 < N_HID) v.y = W2[(k0 + 1) * N_OUT + n];
        }
        B2c[c] = v;
    }

    // ---- Recurrent state in WMMA C/D layout: lane=(n,h), VGPR r -> row m = r + 8h
    v8f mem1 = {};
    v8f mem2 = {};
    float spk1p[8] = {0,0,0,0,0,0,0,0};
    float spk2p[8] = {0,0,0,0,0,0,0,0};

    const size_t O_mem1 = (size_t)N_STEPS * BATCHSZ * N_HID;
    const size_t O_spk2 = 2 * O_mem1;
    const size_t O_mem2 = O_spk2 + (size_t)N_STEPS * BATCHSZ * N_OUT;

    for (int t = 0; t < N_STEPS; ++t) {
        // A1 = x_t tile, 16x4 (K padded 2->4): lanes 0-15 hold (x0,x1), upper half = 0
        v2f A1 = {0.f, 0.f};
        if (h == 0) {
            const float* p = x + ((size_t)t * BATCHSZ + b0 + n) * N_IN;
            A1.x = p[0];
            A1.y = p[1];
            if (t + 1 < N_STEPS) __builtin_prefetch(p + (size_t)BATCHSZ * N_IN, 0, 3);
        }

        // curr1 = x@W1 + b1 (bias in accumulator)
        v8f d1 = __builtin_amdgcn_wmma_f32_16x16x4_f32(false, A1, false, B1,
                                                       (short)0, cbias1, false, false);
        // Leaky integrate + fire, layer 1 (reset-by-subtract uses previous spike)
        float spk1n[8];
        #pragma unroll
        for (int r = 0; r < 8; ++r) {
            float m1 = BETA * mem1[r] + d1[r] - spk1p[r] * THRESH;
            mem1[r]  = m1;
            spk1n[r] = (m1 > THRESH) ? 1.f : 0.f;
        }

        // Stage layer-1 tile images in LDS (also the spk1 transpose source)
        if (n < N_HID) {
            #pragma unroll
            for (int r = 0; r < 8; ++r) {
                const int m = r + 8 * h;
                s_spk1[m * N_HID + n] = spk1n[r];
                s_mem1[m * N_HID + n] = mem1[r];
            }
        }
        __builtin_amdgcn_wave_barrier();   // same-wave DS ops complete in order

        // Layer-2 A chunks (16x4 each) read from the spk1 image: row m = lane's n slot
        const int m = n;
        v2f A20 = *(const v2f*)&s_spk1[m * N_HID + 0 + 2 * h];   // K 0..3
        v2f A21 = *(const v2f*)&s_spk1[m * N_HID + 4 + 2 * h];   // K 4..7
        v2f A22 = {0.f, 0.f};
        if (h == 0) A22 = *(const v2f*)&s_spk1[m * N_HID + 8];   // K 8..9 (10,11 = pad)

        // curr2 = spk1@W2 + b2 (bias in first accumulator)
        v8f d2 = __builtin_amdgcn_wmma_f32_16x16x4_f32(false, A20, false, B2c[0],
                                                       (short)0, cbias2, false, false);
        d2 = __builtin_amdgcn_wmma_f32_16x16x4_f32(false, A21, false, B2c[1],
                                                   (short)0, d2, false, false);
        d2 = __builtin_amdgcn_wmma_f32_16x16x4_f32(false, A22, false, B2c[2],
                                                   (short)0, d2, false, false);

        float spk2n[8];
        #pragma unroll
        for (int r = 0; r < 8; ++r) {
            float m2 = BETA * mem2[r] + d2[r] - spk2p[r] * THRESH;
            mem2[r]  = m2;
            spk2n[r] = (m2 > THRESH) ? 1.f : 0.f;
        }
        if (n < N_OUT) {
            #pragma unroll
            for (int r = 0; r < 8; ++r) {
                const int mm = r + 8 * h;
                s_spk2[mm * N_OUT + n] = spk2n[r];
                s_mem2[mm * N_OUT + n] = mem2[r];
            }
        }
        #pragma unroll
        for (int r = 0; r < 8; ++r) { spk1p[r] = spk1n[r]; spk2p[r] = spk2n[r]; }
        __builtin_amdgcn_wave_barrier();

        // Wide coalesced stores straight from the LDS images:
        // 160-float regions: 1x global_store_b128 (512B/wave) + 1x b32; 64-float: 1x b64.
        const size_t base1 = ((size_t)t * BATCHSZ + b0) * N_HID;   // multiple of 160
        const size_t base2 = ((size_t)t * BATCHSZ + b0) * N_OUT;   // multiple of 64
        float* o_spk1 = out + base1;
        float* o_mem1 = out + O_mem1 + base1;
        float* o_spk2 = out + O_spk2 + base2;
        float* o_mem2 = out + O_mem2 + base2;

        *(v4f*)&o_spk1[lane * 4] = *(const v4f*)&s_spk1[lane * 4];
        o_spk1[128 + lane]       = s_spk1[128 + lane];
        *(v4f*)&o_mem1[lane * 4] = *(const v4f*)&s_mem1[lane * 4];
        o_mem1[128 + lane]       = s_mem1[128 + lane];
        *(v2f*)&o_spk2[lane * 2] = *(const v2f*)&s_spk2[lane * 2];
        *(v2f*)&o_mem2[lane * 2] = *(const v2f*)&s_mem2[lane * 2];
    }
}

extern "C" void kernel_launch(void* const* d_in, const int* in_sizes, int n_in,
                              void* d_out, int out_size, void* d_ws, size_t ws_size,
                              hipStream_t stream) {
    const float* x  = (const float*)d_in[0];
    const float* W1 = (const float*)d_in[1];
    const float* b1 = (const float*)d_in[2];
    const float* W2 = (const float*)d_in[3];
    const float* b2 = (const float*)d_in[4];
    float* out = (float*)d_out;

    dim3 block(128);                 // 4 waves, 16 batch rows each
    dim3 grid(BATCHSZ / 64);         // 1024 blocks -> 4096 waves
    snn_fwd_wmma<<<grid, block, 0, stream>>>(x, W1, b1, W2, b2, out);
}